// CRF_214748365186
// MI455X (gfx1250) — compile-verified
//
#include <hip/hip_runtime.h>

// CRF log-likelihood, MI455X (gfx1250, wave32).
// Forward scan as per-step f32 GEMM vs E=exp(transitions) via
// V_WMMA_F32_16X16X4_F32; B-fragments register-resident across the scan;
// previous alpha kept in registers (D layout) for the masked blend; LDS used
// only for the D->A fragment transpose; emission tiles streamed by the
// Tensor Data Mover (double-buffered, s_wait_tensorcnt) when available.

#define B_ 512
#define S_ 1024
#define T_ 48

typedef float v2f __attribute__((ext_vector_type(2)));
typedef float v8f __attribute__((ext_vector_type(8)));
typedef unsigned int u32x4 __attribute__((ext_vector_type(4)));
typedef int i32x4 __attribute__((ext_vector_type(4)));
typedef int i32x8 __attribute__((ext_vector_type(8)));

#if defined(__has_builtin)
#if __has_builtin(__builtin_amdgcn_tensor_load_to_lds) && \
    __has_builtin(__builtin_amdgcn_s_wait_tensorcnt)
#define HAVE_TDM 1
#endif
#endif
#ifndef HAVE_TDM
#define HAVE_TDM 0
#endif

#if HAVE_TDM
#warning "CDNA5 marker: TDM path ACTIVE (tensor_load_to_lds)"
#else
#warning "CDNA5 marker: TDM path INACTIVE (float4 fallback)"
#endif

#if HAVE_TDM
// Issue a TDM 2D tile load: 16 rows x 48 f32, row stride S_*T_ elements,
// global -> LDS at byte offset lds_off. Tracked by TENSORcnt.
__device__ __forceinline__ void tdm_load_emit_tile(const float* gsrc,
                                                   unsigned lds_off) {
  unsigned long long ga = (unsigned long long)gsrc;
  u32x4 g0;
  g0[0] = 1u;                                    // count=1 (valid descriptor)
  g0[1] = lds_off;                               // LDS dest byte address
  g0[2] = (unsigned)(ga & 0xFFFFFFFFu);          // global addr [31:0]
  g0[3] = (unsigned)((ga >> 32) & 0x01FFFFFFu)   // global addr [56:32]
        | (2u << 30);                            // type = 2 ("image")
  i32x8 g1;
  g1[0] = (int)(2u << 16);                       // data_size = 4 bytes
  g1[1] = (int)((unsigned)T_ << 16);             // tensor_dim0 = 48
  g1[2] = (int)(16u << 16);                      // tensor_dim1 = 16
  g1[3] = (int)((unsigned)T_ << 16);             // tile_dim0 = 48
  g1[4] = 16;                                    // tile_dim1 = 16, tile_dim2 = 0
  g1[5] = (int)(S_ * T_);                        // tensor_dim0_stride (elems)
  g1[6] = 0;
  g1[7] = 0;
  i32x4 gz = {0, 0, 0, 0};
#if __has_include(<hip/amd_detail/amd_gfx1250_TDM.h>)
  i32x8 gz8 = {0, 0, 0, 0, 0, 0, 0, 0};
  __builtin_amdgcn_tensor_load_to_lds(g0, g1, gz, gz, gz8, 0);   // clang-23 lane
#else
  __builtin_amdgcn_tensor_load_to_lds(g0, g1, gz, gz, 0);        // ROCm 7.2 lane
#endif
}
#endif

__global__ __launch_bounds__(32) void crf_forward_kernel(
    const float* __restrict__ inputs,   // [B,S,T]
    const int*   __restrict__ mask,     // [B,S]
    const float* __restrict__ trans,    // [T,T]
    const float* __restrict__ startT,   // [T]
    const float* __restrict__ endT,     // [T]
    float*       __restrict__ logZ)     // [B]
{
  __shared__ float Ee[T_ * T_];         // exp(transitions)
  __shared__ float alpha[16][T_];       // scan state (for D->A transpose)
  __shared__ float emitBuf[2][16 * T_]; // double-buffered emission tiles

  const int lane = threadIdx.x;         // wave32
  const int b0   = blockIdx.x * 16;
  const int r    = lane >> 1;           // staging row (init / fallback)
  const int h    = lane & 1;
  const int arow = lane & 15;           // WMMA half-wave lane index
  const int asel = lane >> 4;           // WMMA half select

  for (int idx = lane; idx < T_ * T_; idx += 32)
    Ee[idx] = __expf(trans[idx]);

  // alpha0 = start_transitions + logits[t=0]
  {
    const float* src = inputs + ((size_t)(b0 + r) * S_) * T_ + h * 24;
    #pragma unroll
    for (int k = 0; k < 24; ++k)
      alpha[r][h * 24 + k] = startT[h * 24 + k] + src[k];
  }
  __syncthreads();

  // Register-resident, loop-invariant B fragments of E (3 N-tiles x 12 K-chunks)
  v2f bfrag[3][12];
  #pragma unroll
  for (int nt = 0; nt < 3; ++nt)
    #pragma unroll
    for (int kc = 0; kc < 12; ++kc) {
      const int k0 = 4 * kc + 2 * asel;
      bfrag[nt][kc].x = Ee[(k0    ) * T_ + nt * 16 + arow];
      bfrag[nt][kc].y = Ee[(k0 + 1) * T_ + nt * 16 + arow];
    }

  // Register copy of alpha in D (accumulator) layout for the masked blend
  float aprev[3][8];
  #pragma unroll
  for (int nt = 0; nt < 3; ++nt)
    #pragma unroll
    for (int v = 0; v < 8; ++v)
      aprev[nt][v] = alpha[v + 8 * asel][nt * 16 + arow];

#if HAVE_TDM
  // Kick off the first emission tile (t = 1) into buffer 1.
  tdm_load_emit_tile(inputs + ((size_t)b0 * S_ + 1) * T_,
                     (unsigned)(unsigned long long)&emitBuf[1][0]);
#endif

  #pragma unroll 1
  for (int t = 1; t < S_; ++t) {
    const float* curEmit = &emitBuf[t & 1][0];

#if HAVE_TDM
    if (t + 1 < S_) {
      tdm_load_emit_tile(inputs + ((size_t)b0 * S_ + (t + 1)) * T_,
                         (unsigned)(unsigned long long)&emitBuf[(t + 1) & 1][0]);
      __builtin_amdgcn_s_wait_tensorcnt(1);   // tile(t) landed; tile(t+1) in flight
    } else {
      __builtin_amdgcn_s_wait_tensorcnt(0);
    }
    asm volatile("" ::: "memory");            // keep emit ds_loads below the wait
#else
    {
      const float* src = inputs + ((size_t)(b0 + r) * S_ + t) * T_ + h * 24;
      #pragma unroll
      for (int k = 0; k < 24; k += 4)
        *(float4*)&emitBuf[t & 1][r * T_ + h * 24 + k] = *(const float4*)(src + k);
    }
    if (t + 1 < S_)
      __builtin_prefetch(inputs + ((size_t)(b0 + r) * S_ + (t + 1)) * T_ + h * 24, 0, 0);
    __syncthreads();
#endif

    float mval = 0.f;
    if (lane < 16) mval = (float)mask[(size_t)(b0 + lane) * S_ + t];

    // alpha read directly in A-fragment layout (12 x ds_load_b64)
    v2f av[12];
    #pragma unroll
    for (int kc = 0; kc < 12; ++kc)
      av[kc] = *(const v2f*)&alpha[arow][4 * kc + 2 * asel];

    // row max across both half-wave owners, then exp into A fragments
    float mx = -3.402823466e38f;
    #pragma unroll
    for (int kc = 0; kc < 12; ++kc) mx = fmaxf(mx, fmaxf(av[kc].x, av[kc].y));
    mx = fmaxf(mx, __shfl_xor(mx, 16, 32));
    v2f afrag[12];
    #pragma unroll
    for (int kc = 0; kc < 12; ++kc) {
      afrag[kc].x = __expf(av[kc].x - mx);
      afrag[kc].y = __expf(av[kc].y - mx);
    }

    // broadcast rowmax / mask to this lane's D-layout rows (rows live in lanes 0..15)
    float mrowv[8], mkv[8];
    #pragma unroll
    for (int v = 0; v < 8; ++v) {
      const int mr = v + 8 * asel;
      mrowv[v] = __shfl(mx,   mr, 32);
      mkv[v]   = __shfl(mval, mr, 32);
    }

    // sumexp(16x48) = u @ E via f32 WMMA; fold in emit + m + log, masked blend
    #pragma unroll
    for (int nt = 0; nt < 3; ++nt) {
      v8f acc = {0.f, 0.f, 0.f, 0.f, 0.f, 0.f, 0.f, 0.f};
      #pragma unroll
      for (int kc = 0; kc < 12; ++kc)
        acc = __builtin_amdgcn_wmma_f32_16x16x4_f32(
            false, afrag[kc], false, bfrag[nt][kc], (short)0, acc, false, false);
      #pragma unroll
      for (int v = 0; v < 8; ++v) {
        const int mr = v + 8 * asel;
        const int n  = nt * 16 + arow;
        float na  = curEmit[mr * T_ + n] + mrowv[v] + __logf(acc[v]);
        float nv  = na * mkv[v] + aprev[nt][v] * (1.0f - mkv[v]);
        aprev[nt][v] = nv;        // register copy for next blend
        alpha[mr][n] = nv;        // LDS copy for next step's A-layout read
      }
    }
    // wave-private DS ops are in-order: next iteration's alpha loads see these stores
  }
  __syncthreads();

  // log_denominator = logsumexp(alpha + end_transitions)
  if (lane < 16) {
    float mx = -3.402823466e38f;
    for (int k = 0; k < T_; ++k) mx = fmaxf(mx, alpha[lane][k] + endT[k]);
    float s = 0.f;
    for (int k = 0; k < T_; ++k) s += __expf(alpha[lane][k] + endT[k] - mx);
    logZ[b0 + lane] = mx + __logf(s);
  }
}

// Numerator: gold-path score, one thread per batch element.
__global__ __launch_bounds__(256) void crf_score_kernel(
    const float* __restrict__ inputs, const int* __restrict__ tags,
    const int* __restrict__ mask, const float* __restrict__ trans,
    const float* __restrict__ startT, const float* __restrict__ endT,
    float* __restrict__ scoreOut)
{
  int b = blockIdx.x * blockDim.x + threadIdx.x;
  if (b >= B_) return;
  const int*   tg = tags  + (size_t)b * S_;
  const int*   mb = mask  + (size_t)b * S_;
  const float* lg = inputs + (size_t)b * S_ * T_;

  int   prev  = tg[0];
  int   msum  = mb[0];
  float score = startT[prev];
  #pragma unroll 1
  for (int t = 0; t + 1 < S_; ++t) {
    int cur = tg[t + 1];
    score += trans[prev * T_ + cur] * (float)mb[t + 1]
           + lg[(size_t)t * T_ + prev] * (float)mb[t];
    msum += mb[t + 1];
    prev = cur;
  }
  int last_idx = msum - 1;
  int last_tag = tg[last_idx];
  score += endT[last_tag] + lg[(size_t)(S_ - 1) * T_ + last_tag] * (float)mb[S_ - 1];
  scoreOut[b] = score;
}

// Deterministic final reduction: sum(score - logZ) over B=512.
__global__ __launch_bounds__(256) void crf_reduce_kernel(
    const float* __restrict__ logZ, const float* __restrict__ score,
    float* __restrict__ out)
{
  __shared__ float sh[256];
  int tid = threadIdx.x;
  sh[tid] = (score[tid] - logZ[tid]) + (score[tid + 256] - logZ[tid + 256]);
  __syncthreads();
  for (int s = 128; s > 0; s >>= 1) {
    if (tid < s) sh[tid] += sh[tid + s];
    __syncthreads();
  }
  if (tid == 0) out[0] = sh[0];
}

extern "C" void kernel_launch(void* const* d_in, const int* in_sizes, int n_in,
                              void* d_out, int out_size, void* d_ws, size_t ws_size,
                              hipStream_t stream) {
  const float* inputs = (const float*)d_in[0];  // [B,S,T] f32
  const int*   tags   = (const int*)  d_in[1];  // [B,S]   i32
  const int*   mask   = (const int*)  d_in[2];  // [B,S]   i32
  const float* trans  = (const float*)d_in[3];  // [T,T]   f32
  const float* startT = (const float*)d_in[4];  // [T]     f32
  const float* endT   = (const float*)d_in[5];  // [T]     f32
  float* out   = (float*)d_out;                 // scalar f32
  float* logZ  = (float*)d_ws;                  // [B]
  float* score = logZ + B_;                     // [B]

  crf_forward_kernel<<<B_ / 16, 32, 0, stream>>>(inputs, mask, trans, startT, endT, logZ);
  crf_score_kernel<<<B_ / 256, 256, 0, stream>>>(inputs, tags, mask, trans, startT, endT, score);
  crf_reduce_kernel<<<1, 256, 0, stream>>>(logZ, score, out);
}